// EmbeddingLoss_22187801051866
// MI455X (gfx1250) — compile-verified
//
#include <hip/hip_runtime.h>

typedef float v2f __attribute__((ext_vector_type(2)));
typedef float v8f __attribute__((ext_vector_type(8)));

#define BATCH 4
#define NPIX  4096          // W*H = 64*64 (also ch*w due to raw reshape)
#define CH    64
#define TPR   (NPIX / 16)   // 256 tiles per row
#define TPB   (TPR * TPR)   // 65536 tiles per batch
#define WAVES 8             // waves per block (256 threads, wave32)
#define TOTAL_TILES (BATCH * TPB)       // 262144
#define NBLOCKS (TOTAL_TILES / WAVES)   // 32768
#define MARGIN 0.5f

// ---------------------------------------------------------------------------
// Kernel 1: pw[b,n] = 1 / (#pixels in batch b with same label as pixel n)
// ---------------------------------------------------------------------------
__global__ void el_count_kernel(const int* __restrict__ lab,
                                float* __restrict__ pw) {
    __shared__ int sl[NPIX];
    const int b = blockIdx.x;
    const int* L = lab + (size_t)b * NPIX;
    for (int t = threadIdx.x; t < NPIX; t += blockDim.x) sl[t] = L[t];
    __syncthreads();
    for (int n = threadIdx.x; n < NPIX; n += blockDim.x) {
        const int ln = sl[n];
        int cnt = 0;
        #pragma unroll 8
        for (int m = 0; m < NPIX; ++m) cnt += (sl[m] == ln) ? 1 : 0;
        pw[(size_t)b * NPIX + n] = 1.0f / (float)cnt;
    }
}

// ---------------------------------------------------------------------------
// Kernel 2: one wave computes one 16x16 tile of sim = E * E^T via
// V_WMMA_F32_16X16X4_F32 (K=64 -> 16 chained WMMAs), applies the weighted
// contrastive loss elementwise, reduces to one partial per block.
// ---------------------------------------------------------------------------
__global__ void el_tile_kernel(const float* __restrict__ emb,
                               const int* __restrict__ lab,
                               const float* __restrict__ pw,
                               float* __restrict__ partials) {
    const int lane = threadIdx.x & 31;
    const int wave = threadIdx.x >> 5;

    // Global tile id -> (batch, tile-row, tile-col). Fully uniform per wave.
    const unsigned tile = (unsigned)blockIdx.x * WAVES + wave;
    const unsigned b  = tile >> 16;            // tile / TPB
    const unsigned t  = tile & (TPB - 1);
    const unsigned ti = t >> 8;                // tile row (0..255)
    const unsigned tj = t & (TPR - 1);         // tile col (0..255)
    const int i0 = (int)ti * 16;
    const int j0 = (int)tj * 16;

    const float* Eb = emb + (size_t)b * NPIX * CH;

    // f32 16x4 A-fragment layout: lane half h=lane/16 selects K pair {2h,2h+1},
    // VGPR within the float2 selects element of the pair; M = lane%16.
    // B (4x16) fragment for E^T is loaded with the identical pattern with j0.
    const int h = lane >> 4;      // 0 or 1
    const int m = lane & 15;
    const float* arow = Eb + (size_t)(i0 + m) * CH + 2 * h;  // 8B aligned
    const float* brow = Eb + (size_t)(j0 + m) * CH + 2 * h;

    v8f c = {};
    #pragma unroll
    for (int k = 0; k < CH / 4; ++k) {
        v2f a  = *(const v2f*)(arow + 4 * k);
        v2f bb = *(const v2f*)(brow + 4 * k);
        // 8 args: (neg_a, A, neg_b, B, c_mod, C, reuse_a, reuse_b)
        c = __builtin_amdgcn_wmma_f32_16x16x4_f32(
                false, a, false, bb, (short)0, c, false, false);
    }

    // Elementwise loss. C/D layout: vgpr r, lane -> M = r + 8*h, N = lane%16.
    const int*   Lb  = lab + (size_t)b * NPIX;
    const float* Pb  = pw  + (size_t)b * NPIX;
    const int   j   = j0 + m;
    const int   lj  = Lb[j];
    const float pwj = Pb[j];
    const int   ibase = i0 + 8 * h;

    float acc = 0.0f;
    #pragma unroll
    for (int r = 0; r < 8; ++r) {
        const int   i   = ibase + r;
        const int   li  = Lb[i];
        const float pwi = Pb[i];
        const float s   = c[r];
        const float v   = (li == lj) ? (1.0f - s) : fmaxf(s - MARGIN, 0.0f);
        acc += pwi * pwj * v;
    }

    // Wave32 reduction
    #pragma unroll
    for (int off = 16; off > 0; off >>= 1) acc += __shfl_xor(acc, off, 32);

    __shared__ float sred[WAVES];
    if (lane == 0) sred[wave] = acc;
    __syncthreads();
    if (threadIdx.x == 0) {
        float s = 0.0f;
        #pragma unroll
        for (int w = 0; w < WAVES; ++w) s += sred[w];
        partials[blockIdx.x] = s;
    }
}

// ---------------------------------------------------------------------------
// Kernel 3: deterministic fixed-order reduction of block partials; apply the
// final *N scale from the reference (total = N * sum_b S_b).
// ---------------------------------------------------------------------------
__global__ void el_reduce_kernel(const float* __restrict__ partials,
                                 int np, float* __restrict__ out) {
    __shared__ float sm[256];
    float acc = 0.0f;
    for (int i = threadIdx.x; i < np; i += 256) acc += partials[i];
    sm[threadIdx.x] = acc;
    __syncthreads();
    for (int s = 128; s > 0; s >>= 1) {
        if (threadIdx.x < s) sm[threadIdx.x] += sm[threadIdx.x + s];
        __syncthreads();
    }
    if (threadIdx.x == 0) out[0] = (float)NPIX * sm[0];
}

// ---------------------------------------------------------------------------
extern "C" void kernel_launch(void* const* d_in, const int* in_sizes, int n_in,
                              void* d_out, int out_size, void* d_ws, size_t ws_size,
                              hipStream_t stream) {
    const float* emb = (const float*)d_in[0];   // [4,64,64,64] f32, contiguous
    const int*   lab = (const int*)d_in[1];     // [4,64,64] i32

    float* pw       = (float*)d_ws;             // BATCH*NPIX floats (64 KB)
    float* partials = pw + (size_t)BATCH * NPIX; // NBLOCKS floats (128 KB)

    el_count_kernel<<<BATCH, 256, 0, stream>>>(lab, pw);
    el_tile_kernel<<<NBLOCKS, WAVES * 32, 0, stream>>>(emb, lab, pw, partials);
    el_reduce_kernel<<<1, 256, 0, stream>>>(partials, NBLOCKS, (float*)d_out);
}